// StressGNN_46608985096656
// MI455X (gfx1250) — compile-verified
//
#include <hip/hip_runtime.h>

// ---------------------------------------------------------------------------
// GCN 2-layer forward for MI455X (gfx1250, wave32).
//  - Dense GEMMs (x@W) use V_WMMA_F32_16X16X4_F32 (full f32 precision).
//  - Edge aggregation = coalesced float4 gathers + global f32 L2 atomics
//    (exact scatter-add semantics of the reference; everything L2-resident).
// ---------------------------------------------------------------------------

typedef __attribute__((ext_vector_type(2))) float v2f;
typedef __attribute__((ext_vector_type(8))) float v8f;

#define GCN_N_NODES 100000

// ---------------- utility: fill buffer with a constant --------------------
__global__ void gcn_fill_kernel(float* __restrict__ p, float v, int n) {
    int i = blockIdx.x * blockDim.x + threadIdx.x;
    if (i < n) p[i] = v;
}

// ---------------- degree count via float atomics (deg pre-filled to 1.0 for
// the self loop; counts are exact in f32 far below 2^24) --------------------
__global__ void gcn_degree_kernel(const int* __restrict__ dst,
                                  float* __restrict__ deg, int nE) {
    int e = blockIdx.x * blockDim.x + threadIdx.x;
    if (e < nE) atomicAdd(&deg[dst[e]], 1.0f);
}

// ---------------- deg -> deg^-1/2 in place --------------------------------
__global__ void gcn_rsqrt_kernel(float* __restrict__ deg, int n) {
    int i = blockIdx.x * blockDim.x + threadIdx.x;
    if (i < n) deg[i] = rsqrtf(deg[i]);
}

// ---------------- dense GEMM H = X @ W via f32 WMMA ------------------------
// X: [nRows x K] row-major, W: [K x F] row-major, H: [nRows x F].
// nRows % 16 == 0, K % 4 == 0, F % 16 == 0 (all true here).
// One wave computes one 16x16 output tile; 8 waves per 256-thread block.
// Whole waves exit early only => EXEC is all-ones at every WMMA.
__global__ void gcn_gemm_wmma_f32(const float* __restrict__ X,
                                  const float* __restrict__ W,
                                  float* __restrict__ H,
                                  int nRows, int K, int F) {
    const int wavesPerBlock = blockDim.x >> 5;
    const int waveId = blockIdx.x * wavesPerBlock + (threadIdx.x >> 5);
    const int nTilesN = F >> 4;
    const int totalWaves = (nRows >> 4) * nTilesN;
    if (waveId >= totalWaves) return;   // whole wave exits together

    const int tileM = waveId / nTilesN;
    const int tileN = waveId % nTilesN;
    const int lane  = threadIdx.x & 31;
    const int half  = lane >> 4;        // 0: K pair {0,1}, 1: K pair {2,3}
    const int l15   = lane & 15;

    const int rowA = tileM * 16 + l15;  // A-matrix row handled by this lane
    const int colB = tileN * 16 + l15;  // B/C/D column handled by this lane

    const float* __restrict__ xrow = X + (size_t)rowA * K;

    v8f c = {};
    for (int k0 = 0; k0 < K; k0 += 4) {
        const int ka = k0 + 2 * half;
        v2f a, b;
        // A 16x4 f32 fragment: VGPR0 = K=0/2, VGPR1 = K=1/3 (ISA 7.12.2)
        a[0] = xrow[ka];
        a[1] = xrow[ka + 1];
        // B 4x16 f32 fragment: lanes index N, halves index K pairs
        b[0] = W[(size_t)ka * F + colB];
        b[1] = W[(size_t)(ka + 1) * F + colB];
        // D = A*B + C   (8-arg form: neg_a, A, neg_b, B, c_mod, C, reuseA, reuseB)
        c = __builtin_amdgcn_wmma_f32_16x16x4_f32(
                false, a, false, b, (short)0, c, false, false);
    }

    // C/D layout: VGPR r holds row M = r + 8*half, column = colB
    const int rowD0 = tileM * 16 + 8 * half;
#pragma unroll
    for (int r = 0; r < 8; ++r)
        H[(size_t)(rowD0 + r) * F + colB] = c[r];
}

// ---------------- edge aggregation: agg[dst] += norm * h[src] --------------
// F/4 threads per edge; each thread gathers a float4 (GLOBAL_LOAD_B128)
// and issues 4 f32 atomics that resolve in L2.
template <int F>
__global__ void gcn_edge_agg_kernel(const int* __restrict__ src,
                                    const int* __restrict__ dst,
                                    const float* __restrict__ dinv,
                                    const float* __restrict__ h,
                                    float* __restrict__ agg, int nE) {
    constexpr int TPE = F / 4;          // threads per edge
    constexpr int EPB = 256 / TPE;      // edges per 256-thread block
    const int t  = threadIdx.x;
    const int e  = blockIdx.x * EPB + t / TPE;
    const int f4 = (t % TPE) * 4;
    if (e >= nE) return;

    const int s = src[e];
    const int d = dst[e];
    const float w = dinv[s] * dinv[d];

    const float4 hv = *reinterpret_cast<const float4*>(h + (size_t)s * F + f4);
    float* ap = agg + (size_t)d * F + f4;
    atomicAdd(ap + 0, w * hv.x);
    atomicAdd(ap + 1, w * hv.y);
    atomicAdd(ap + 2, w * hv.z);
    atomicAdd(ap + 3, w * hv.w);
}

// ---------------- self-loop + bias + ReLU (in place on agg) ----------------
__global__ void gcn_self_bias_relu(const float* __restrict__ hraw,
                                   const float* __restrict__ dinv,
                                   const float* __restrict__ bias,
                                   float* __restrict__ agg,
                                   int nRows, int F) {
    const int i = blockIdx.x * blockDim.x + threadIdx.x;
    if (i >= nRows * F) return;
    const int node = i / F;
    const int f = i - node * F;
    const float di = dinv[node];
    const float v = agg[i] + di * di * hraw[i] + bias[f];
    agg[i] = v > 0.0f ? v : 0.0f;
}

// ---------------- column sums of h2 (per-block accumulate + atomics) -------
__global__ void gcn_colsum_kernel(const float* __restrict__ h,
                                  float* __restrict__ g, int nRows) {
    const int f = threadIdx.x;          // blockDim.x == 128
    float s = 0.0f;
    for (int r = blockIdx.x; r < nRows; r += gridDim.x)
        s += h[(size_t)r * 128 + f];
    atomicAdd(&g[f], s);
}

// ---------------- final: out = (g/N) @ Wfc + bfc ---------------------------
__global__ void gcn_final_kernel(const float* __restrict__ g,
                                 const float* __restrict__ Wfc,
                                 const float* __restrict__ bfc,
                                 float* __restrict__ out) {
    __shared__ float red[128];
    const int f = threadIdx.x;          // blockDim.x == 128
    red[f] = g[f] * (1.0f / (float)GCN_N_NODES) * Wfc[f];
    __syncthreads();
    for (int s = 64; s > 0; s >>= 1) {
        if (f < s) red[f] += red[f + s];
        __syncthreads();
    }
    if (f == 0) out[0] = red[0] + bfc[0];
}

// ---------------------------------------------------------------------------
extern "C" void kernel_launch(void* const* d_in, const int* in_sizes, int n_in,
                              void* d_out, int out_size, void* d_ws, size_t ws_size,
                              hipStream_t stream) {
    const float* x   = (const float*)d_in[0];
    const int*   ei  = (const int*)  d_in[1];
    const float* W1  = (const float*)d_in[2];
    const float* b1  = (const float*)d_in[3];
    const float* W2  = (const float*)d_in[4];
    const float* b2  = (const float*)d_in[5];
    const float* Wfc = (const float*)d_in[6];
    const float* bfc = (const float*)d_in[7];
    float* out = (float*)d_out;

    const int N  = GCN_N_NODES;
    const int nE = in_sizes[1] / 2;     // edge_index is [2, E]
    const int* src = ei;
    const int* dst = ei + nE;

    // ---- workspace layout (floats), 256B-aligned regions ----
    char* ws = (char*)d_ws;
    size_t off = 0;
    auto alloc = [&](size_t nFloats) {
        float* p = (float*)(ws + off);
        off += ((nFloats * sizeof(float)) + 255) & ~(size_t)255;
        return p;
    };
    float* dinv  = alloc(N);        // deg, then deg^-1/2 in place
    float* g     = alloc(128);      // column sums
    float* h1raw = alloc((size_t)N * 64);   // x @ W1
    float* h1    = alloc((size_t)N * 64);   // agg1 -> relu'd layer-1 output
    float* h2raw = alloc((size_t)N * 128);  // h1 @ W2
    float* h2    = alloc((size_t)N * 128);  // agg2 -> relu'd layer-2 output
    (void)ws_size; (void)n_in; (void)out_size;

    // ---- degrees: deg = 1 (self loop) + count(dst); dinv = deg^-1/2 ----
    gcn_fill_kernel<<<(N + 255) / 256, 256, 0, stream>>>(dinv, 1.0f, N);
    gcn_degree_kernel<<<(nE + 255) / 256, 256, 0, stream>>>(dst, dinv, nE);
    gcn_rsqrt_kernel<<<(N + 255) / 256, 256, 0, stream>>>(dinv, N);

    // ---- layer 1: h1raw = x @ W1 (WMMA), aggregate, relu ----
    {
        const int waves = (N / 16) * (64 / 16);             // 25000
        gcn_gemm_wmma_f32<<<(waves + 7) / 8, 256, 0, stream>>>(x, W1, h1raw, N, 8, 64);
        const int n1 = N * 64;
        gcn_fill_kernel<<<(n1 + 255) / 256, 256, 0, stream>>>(h1, 0.0f, n1);
        const int epb = 256 / (64 / 4);                     // 16 edges/block
        gcn_edge_agg_kernel<64><<<(nE + epb - 1) / epb, 256, 0, stream>>>(
            src, dst, dinv, h1raw, h1, nE);
        gcn_self_bias_relu<<<(n1 + 255) / 256, 256, 0, stream>>>(
            h1raw, dinv, b1, h1, N, 64);
    }

    // ---- layer 2: h2raw = h1 @ W2 (WMMA), aggregate, relu ----
    {
        const int waves = (N / 16) * (128 / 16);            // 50000
        gcn_gemm_wmma_f32<<<(waves + 7) / 8, 256, 0, stream>>>(h1, W2, h2raw, N, 64, 128);
        const int n2 = N * 128;
        gcn_fill_kernel<<<(n2 + 255) / 256, 256, 0, stream>>>(h2, 0.0f, n2);
        const int epb = 256 / (128 / 4);                    // 8 edges/block
        gcn_edge_agg_kernel<128><<<(nE + epb - 1) / epb, 256, 0, stream>>>(
            src, dst, dinv, h2raw, h2, nE);
        gcn_self_bias_relu<<<(n2 + 255) / 256, 256, 0, stream>>>(
            h2raw, dinv, b2, h2, N, 128);
    }

    // ---- readout: g = colsum(h2); out = (g/N) @ Wfc + bfc ----
    gcn_fill_kernel<<<1, 128, 0, stream>>>(g, 0.0f, 128);
    gcn_colsum_kernel<<<256, 128, 0, stream>>>(h2, g, N);
    gcn_final_kernel<<<1, 128, 0, stream>>>(g, Wfc, bfc, out);
}